// Encoder_79001628443245
// MI455X (gfx1250) — compile-verified
//
#include <hip/hip_runtime.h>
#include <stdint.h>

#define N_GENES 20000
#define N_DRUGS 4000
#define IN_C    1024
#define OUT_C   512
#define N_EDGES 262144

#define MTILES_PER_WAVE 5            // 5 x 16 = 80 rows of xw per wave
#define WAVES_PER_BLOCK 4            // 4 N-tiles (64 cols) per block

typedef __attribute__((ext_vector_type(2))) float v2f;
typedef __attribute__((ext_vector_type(8))) float v8f;

// ---------------------------------------------------------------------------
// 1) zero degree arrays
// ---------------------------------------------------------------------------
__global__ void zero_deg_kernel(float* __restrict__ row_deg,
                                float* __restrict__ col_deg) {
    int i = blockIdx.x * blockDim.x + threadIdx.x;
    if (i < N_GENES) row_deg[i] = 0.0f;
    if (i < N_DRUGS) col_deg[i] = 0.0f;
}

// ---------------------------------------------------------------------------
// 2) degree counting (segment_sum of ones) via f32 atomics (L2-resident)
// ---------------------------------------------------------------------------
__global__ void deg_kernel(const long long* __restrict__ src,
                           const long long* __restrict__ dst,
                           float* __restrict__ row_deg,
                           float* __restrict__ col_deg) {
    int e = blockIdx.x * blockDim.x + threadIdx.x;
    if (e >= N_EDGES) return;
    atomicAdd(&row_deg[(int)src[e]], 1.0f);
    atomicAdd(&col_deg[(int)dst[e]], 1.0f);
}

// ---------------------------------------------------------------------------
// 3) per-edge normalization coefficient.
//    Gathered degrees are always >= 1 (the edge itself), no zero guard needed.
// ---------------------------------------------------------------------------
__global__ void coef_kernel(const long long* __restrict__ src,
                            const long long* __restrict__ dst,
                            const float* __restrict__ row_deg,
                            const float* __restrict__ col_deg,
                            float* __restrict__ coef) {
    int e = blockIdx.x * blockDim.x + threadIdx.x;
    if (e >= N_EDGES) return;
    float rd = row_deg[(int)src[e]];
    float cd = col_deg[(int)dst[e]];
    coef[e] = rsqrtf(rd * cd);
}

// ---------------------------------------------------------------------------
// 4) xw = x @ W with V_WMMA_F32_16X16X4_F32.
//    Block = 128 threads = 4 waves; each wave computes an 80x16 tile of xw
//    (5 M-subtiles sharing every B operand load -> 1 B load : 5 WMMAs).
//    A-layout (16x4 f32, 2 VGPRs): lane<16 -> (K=0,K=1), lane>=16 -> (K=2,K=3)
//      of row M = lane%16  ==> contiguous float2 (b64) load from row-major x;
//      the 5 subtile rows are 64 KiB apart -> immediate-offset loads.
//    B-layout (4x16 f32, 2 VGPRs): VGPR0 = rows K=0 (lanes 0-15) / K=2
//      (lanes 16-31); VGPR1 = rows K=1 / K=3, column N = lane%16.
//    C/D: VGPR v -> row M = v + 8*(lane>=16), col N = lane%16.
// ---------------------------------------------------------------------------
__global__ void __launch_bounds__(128)
gemm_wmma_f32_kernel(const float* __restrict__ x,
                     const float* __restrict__ w,
                     float* __restrict__ xw) {
    const int wave = threadIdx.x >> 5;
    const int lane = threadIdx.x & 31;
    const int hi   = lane >> 4;      // 0: lanes 0-15, 1: lanes 16-31
    const int l16  = lane & 15;

    const int m0 = blockIdx.y * (16 * MTILES_PER_WAVE);
    const int n0 = (blockIdx.x * WAVES_PER_BLOCK + wave) * 16;

    const float* arow = x + (size_t)(m0 + l16) * IN_C + 2 * hi;      // K-pair base
    const float* bcol = w + (size_t)(2 * hi) * OUT_C + n0 + l16;     // K-row base

    v8f acc[MTILES_PER_WAVE];
#pragma unroll
    for (int i = 0; i < MTILES_PER_WAVE; ++i) acc[i] = (v8f){};

#pragma unroll 2
    for (int k = 0; k < IN_C; k += 4) {
        v2f b;
        b.x = bcol[(size_t)k * OUT_C];           // B[k + 2*hi][n]
        b.y = bcol[(size_t)k * OUT_C + OUT_C];   // B[k + 2*hi + 1][n]
#pragma unroll
        for (int i = 0; i < MTILES_PER_WAVE; ++i) {
            v2f a;
            a.x = arow[(size_t)i * 16 * IN_C + k];       // A[m + 16i][k + 2*hi]
            a.y = arow[(size_t)i * 16 * IN_C + k + 1];   // merges to b64
            acc[i] = __builtin_amdgcn_wmma_f32_16x16x4_f32(
                /*neg_a=*/false, a, /*neg_b=*/false, b,
                /*c_mod=*/(short)0, acc[i], /*reuse_a=*/false, /*reuse_b=*/false);
        }
    }

#pragma unroll
    for (int i = 0; i < MTILES_PER_WAVE; ++i) {
        float* orow = xw + (size_t)(m0 + i * 16 + 8 * hi) * OUT_C + n0 + l16;
#pragma unroll
        for (int v = 0; v < 8; ++v)
            orow[(size_t)v * OUT_C] = acc[i][v];
    }
}

// ---------------------------------------------------------------------------
// 5) out[d][c] = bias[c]  (scatter then accumulates on top)
// ---------------------------------------------------------------------------
__global__ void bias_init_kernel(const float* __restrict__ bias,
                                 float* __restrict__ out) {
    int i = blockIdx.x * blockDim.x + threadIdx.x;
    if (i < N_DRUGS * OUT_C) out[i] = bias[i & (OUT_C - 1)];
}

// ---------------------------------------------------------------------------
// 6) out[dst] += coef[e] * xw[src]   (float4 gather + f32 atomics, L2-hot)
//    128 threads per edge (512 ch / 4).
// ---------------------------------------------------------------------------
__global__ void __launch_bounds__(256)
scatter_kernel(const long long* __restrict__ src,
               const long long* __restrict__ dst,
               const float* __restrict__ coef,
               const float* __restrict__ xw,
               float* __restrict__ out) {
    int gid = blockIdx.x * blockDim.x + threadIdx.x;
    int e  = gid >> 7;              // / 128
    int c4 = (gid & 127) << 2;      // channel offset
    if (e >= N_EDGES) return;

    int   s = (int)src[e];
    int   d = (int)dst[e];
    float c = coef[e];

    const float4 v = *reinterpret_cast<const float4*>(xw + (size_t)s * OUT_C + c4);
    float* o = out + (size_t)d * OUT_C + c4;
    atomicAdd(o + 0, v.x * c);
    atomicAdd(o + 1, v.y * c);
    atomicAdd(o + 2, v.z * c);
    atomicAdd(o + 3, v.w * c);
}

// ---------------------------------------------------------------------------
extern "C" void kernel_launch(void* const* d_in, const int* in_sizes, int n_in,
                              void* d_out, int out_size, void* d_ws, size_t ws_size,
                              hipStream_t stream) {
    const float*     x      = (const float*)d_in[0];
    const float*     weight = (const float*)d_in[1];
    const float*     bias   = (const float*)d_in[2];
    const long long* edge   = (const long long*)d_in[3];  // int64 in reference
    const long long* src = edge;              // row 0: gene ids
    const long long* dst = edge + N_EDGES;    // row 1: drug ids
    float* out = (float*)d_out;

    // workspace layout
    char*  ws      = (char*)d_ws;
    float* xw      = (float*)ws;                                  // 40,960,000 B
    float* row_deg = (float*)(ws + (size_t)N_GENES * OUT_C * 4);  // 80,000 B
    float* col_deg = row_deg + N_GENES;                           // 16,000 B
    float* coef    = col_deg + N_DRUGS;                           // 1,048,576 B

    zero_deg_kernel<<<(N_GENES + 255) / 256, 256, 0, stream>>>(row_deg, col_deg);
    deg_kernel<<<(N_EDGES + 255) / 256, 256, 0, stream>>>(src, dst, row_deg, col_deg);
    coef_kernel<<<(N_EDGES + 255) / 256, 256, 0, stream>>>(src, dst, row_deg, col_deg, coef);

    // (8, 250): 4 waves/block over N, 80 rows per wave over M
    dim3 ggrid(OUT_C / (16 * WAVES_PER_BLOCK), N_GENES / (16 * MTILES_PER_WAVE));
    gemm_wmma_f32_kernel<<<ggrid, 128, 0, stream>>>(x, weight, xw);

    bias_init_kernel<<<(N_DRUGS * OUT_C + 255) / 256, 256, 0, stream>>>(bias, out);
    scatter_kernel<<<(N_EDGES * 128) / 256, 256, 0, stream>>>(src, dst, coef, xw, out);
}